// CBOW_33122787787244
// MI455X (gfx1250) — compile-verified
//
#include <hip/hip_runtime.h>
#include <hip/hip_bf16.h>
#include <stdint.h>

// ---------------------------------------------------------------------------
// CBOW gather-sum for MI455X (gfx1250).
// out[b,o] = fc_b[o] + sum_{i<4} fc_w[o, i*8192 + ctx[b,i]]
//
// fc_w (1 GB fp32) is streamed exactly once, row by row (128 KB contiguous),
// into LDS via the Tensor Data Mover with double buffering so each WG always
// has a DMA in flight (HBM roofline: ~1 GB / 23.3 TB/s ~= 43 us).
// The 4-way gather-reduce + bias is done on the matrix pipe with
// V_WMMA_F32_16X16X4_F32 (A = gathered 16x4, B = ones, C = bias splat).
// ---------------------------------------------------------------------------

typedef __attribute__((ext_vector_type(2))) float v2f;
typedef __attribute__((ext_vector_type(8))) float v8f;
typedef __attribute__((ext_vector_type(4))) unsigned int u32x4;
typedef __attribute__((ext_vector_type(8))) int i32x8;
typedef __attribute__((ext_vector_type(4))) int i32x4;

#define VOCAB       8192
#define CTXN        4
#define BATCH       1024
#define ROW_LEN     (CTXN * VOCAB)    // 32768 floats = 128 KB per fc_w row
#define ROWS_PER_WG 8
#define BLOCK       256               // 8 wave32s
#define GROUPS      (BATCH / (BLOCK / 32) / 16)  // 8 WMMA groups per wave

#if __has_builtin(__builtin_amdgcn_tensor_load_to_lds)
#define HAVE_TDM 1
#else
#define HAVE_TDM 0
#endif

__global__ __launch_bounds__(BLOCK)
void cbow_gather_wmma(const int* __restrict__ ctx,
                      const float* __restrict__ w,
                      const float* __restrict__ bias,
                      float* __restrict__ out) {
  __shared__ float row_buf[2][ROW_LEN];   // 2 x 128 KB (double buffer)
  __shared__ int   ctx_s[BATCH * CTXN];   // 16 KB   -> 272 KB total LDS

  const int lane = threadIdx.x & 31;
  const int wave = threadIdx.x >> 5;
  const int m    = lane & 15;             // WMMA A-matrix row (batch in group)
  const int k0   = (lane >> 4) << 1;      // lanes 0-15 -> K=0,1 ; 16-31 -> K=2,3
  const int row_base = blockIdx.x * ROWS_PER_WG;

#if HAVE_TDM
  // Constant part of the TDM descriptor (group 1): one 32768-element 4B row.
  // Bit packing per CDNA5 ISA sec 8.4 (D# group 1).
  i32x8 g1;
  g1[0] = 0x00020000;                 // data_size=2 (4 bytes); no multicast/pad
  g1[1] = (int)0x80000000u;           // tensor_dim0[15:0]=32768 in bits[63:48]
  g1[2] = 0x00010000;                 // tensor_dim0 hi = 0; tensor_dim1 = 1
  g1[3] = (int)0x80000000u;           // tensor_dim1 hi = 0; tile_dim0 = 32768
  g1[4] = 1;                          // tile_dim1 = 1; tile_dim2 = 0
  g1[5] = ROW_LEN;                    // tensor_dim0_stride lo32
  g1[6] = 0;
  g1[7] = 0;
  const i32x4 gz = {0, 0, 0, 0};      // groups 2/3 unused (<=2D tensor)
  const unsigned lds_base =
      (unsigned)(size_t)(&row_buf[0][0]);  // low 32 bits = LDS byte offset

  auto tdm_issue = [&](int r, int buf) {
    const uint64_t ga =
        (uint64_t)(size_t)(w + (size_t)(row_base + r) * ROW_LEN);
    u32x4 g0;
    g0[0] = 1u;                                   // count=1, user descriptor
    g0[1] = lds_base + (unsigned)buf * (ROW_LEN * 4);  // lds_addr (bytes)
    g0[2] = (unsigned)(ga & 0xffffffffu);         // global_addr lo
    g0[3] = (unsigned)((ga >> 32) & 0x01ffffffu)  // global_addr hi (57-bit)
            | 0x80000000u;                        // type = 2 ("image")
#if __clang_major__ >= 23
    const i32x8 gz8 = {0, 0, 0, 0, 0, 0, 0, 0};
    __builtin_amdgcn_tensor_load_to_lds(g0, g1, gz, gz, gz8, 0);
#else
    __builtin_amdgcn_tensor_load_to_lds(g0, g1, gz, gz, 0);
#endif
  };

  // Kick off the first row DMA immediately (overlaps with ctx staging).
  if (wave == 0) tdm_issue(0, 0);
#endif

  // Stage all context indices once (16 KB, L2-resident source).
  for (int t = threadIdx.x; t < BATCH * CTXN; t += BLOCK)
    ctx_s[t] = ctx[t];
  __syncthreads();

  // Hoist row-invariant gather offsets: lane handles batch b = b0+g*16+m,
  // tables k0 and k0+1. Element offsets into a row buffer (in floats).
  const int b0 = wave * (BATCH / (BLOCK / 32));
  int offA[GROUPS], offB[GROUPS];
#pragma unroll
  for (int g = 0; g < GROUPS; ++g) {
    const int b = b0 + g * 16 + m;
    offA[g] = (k0 << 13)       + ctx_s[b * CTXN + k0];
    offB[g] = ((k0 + 1) << 13) + ctx_s[b * CTXN + k0 + 1];
  }

  // All-ones B matrix (4x16 fp32 -> 2 VGPRs, 1.0 in every lane).
  v2f Bones; Bones[0] = 1.0f; Bones[1] = 1.0f;

  for (int r = 0; r < ROWS_PER_WG; ++r) {
#if HAVE_TDM
    if (wave == 0) {
      if (r + 1 < ROWS_PER_WG) {
        // Buffer (r+1)&1 was released by the barrier ending iteration r-1.
        tdm_issue(r + 1, (r + 1) & 1);
        __builtin_amdgcn_s_wait_tensorcnt(1);  // in-order: row r has landed
      } else {
        __builtin_amdgcn_s_wait_tensorcnt(0);
      }
    }
#else
    {  // Fallback: cooperative, fully coalesced float4 copy of the row.
      const float4* __restrict__ src4 =
          (const float4*)(w + (size_t)(row_base + r) * ROW_LEN);
      float4* __restrict__ dst4 = (float4*)&row_buf[r & 1][0];
      for (int t = threadIdx.x; t < ROW_LEN / 4; t += BLOCK)
        dst4[t] = src4[t];
    }
#endif
    __syncthreads();   // row r staged and visible to all waves

    const float* __restrict__ rb = &row_buf[r & 1][0];
    const int o = row_base + r;
    const float bo = bias[o];
    v8f C;
#pragma unroll
    for (int j = 0; j < 8; ++j) C[j] = bo;

#pragma unroll
    for (int g = 0; g < GROUPS; ++g) {
      v2f A;
      A[0] = rb[offA[g]];
      A[1] = rb[offB[g]];
      // D[m,n] = sum_k A[m,k]*1 + bias[o]  (full-EXEC wave, per WMMA rules)
      v8f D = __builtin_amdgcn_wmma_f32_16x16x4_f32(
          false, A, false, Bones, (short)0, C, false, false);
      // Lane 0 holds D[0..7,0] in its 8 acc VGPRs, lane 16 holds D[8..15,0].
      if ((lane & 15) == 0) {
        const int mb = b0 + g * 16 + ((lane >> 4) << 3);
#pragma unroll
        for (int j = 0; j < 8; ++j)
          out[(size_t)(mb + j) * VOCAB + o] = D[j];
      }
    }
    __syncthreads();   // release buffer (r&1) before it is re-targeted
  }
}

extern "C" void kernel_launch(void* const* d_in, const int* in_sizes, int n_in,
                              void* d_out, int out_size, void* d_ws, size_t ws_size,
                              hipStream_t stream) {
  (void)in_sizes; (void)n_in; (void)d_ws; (void)ws_size; (void)out_size;
  const int*   ctx  = (const int*)d_in[0];    // [1024, 4] int32
  const float* fc_w = (const float*)d_in[1];  // [8192, 32768] fp32
  const float* fc_b = (const float*)d_in[2];  // [8192] fp32
  float*       out  = (float*)d_out;          // [1024, 8192] fp32

  dim3 grid(VOCAB / ROWS_PER_WG);             // 1024 workgroups, 8 rows each
  dim3 block(BLOCK);
  cbow_gather_wmma<<<grid, block, 0, stream>>>(ctx, fc_w, fc_b, out);
}